// YOLOLoss_87273735455424
// MI455X (gfx1250) — compile-verified
//
#include <hip/hip_runtime.h>
#include <hip/hip_bf16.h>

typedef __attribute__((ext_vector_type(2))) float v2f;
typedef __attribute__((ext_vector_type(8))) float v8f;

#define NUM_B 16
#define NUM_M 32
#define NUM_A 3
#define NUM_C 80
#define IMG 640.0f

__device__ __forceinline__ float softplusf(float x) {
    // max(x,0) + log1p(exp(-|x|))  == BCE(x, t=0)
    return fmaxf(x, 0.0f) + log1pf(expf(-fabsf(x)));
}

// ws layout (floats):
// [0..2]  S0[s]      : sum softplus over ALL obj cells, per scale
// [3..5]  pos1[s]    : sum (softplus(x)-x) over positive cells (t=1 term)
// [6..8]  possub[s]  : sum softplus(x) over positive cells (to subtract from S0)
// [9..11] cls[s]     : masked cls BCE sum
// [12..14] poscells[s]: # unique positive (b,j,i) cells
// [15]    box_acc    : sum el/denom_b over everything
// [16..31] nv[b]     : # valid boxes per batch

__global__ void k_zero(float* ws) {
    int i = threadIdx.x;
    if (i < 32) ws[i] = 0.0f;
}

// Streaming softplus reduction over the 3 objectness channels of one scale.
// WMMA f32 16x16x4 (B = ones) performs the wave-level reduction in full f32.
__global__ void k_obj(const float* __restrict__ pred, int hw, float* ws, int slot) {
    int hwq = hw >> 2;             // hw divisible by 4 for all scales
    int n4  = NUM_B * NUM_A * hwq;
    float acc = 0.0f;
    for (int i = blockIdx.x * blockDim.x + threadIdx.x; i < n4;
         i += gridDim.x * blockDim.x) {
        int plane = i / hwq;
        int pix4  = i - plane * hwq;
        int b = plane / NUM_A;
        int a = plane - b * NUM_A;
        const float4* p =
            (const float4*)(pred + (size_t)(b * 255 + a * 85 + 4) * hw) + pix4;
        float4 v = *p;
        acc += softplusf(v.x) + softplusf(v.y) + softplusf(v.z) + softplusf(v.w);
    }

    // ---- WMMA f32 wave32 reduction: D = A(16x4 of partials) x B(4x16 ones) ----
    // Lane L<16 supplies A[L][0]; lane L>=16 supplies A[L-16][2]; others zero.
    v2f Am; Am.x = acc;  Am.y = 0.0f;
    v2f Bm; Bm.x = 1.0f; Bm.y = 1.0f;   // all-ones B regardless of layout
    v8f Cm = {};
    v8f Dm = __builtin_amdgcn_wmma_f32_16x16x4_f32(
        false, Am, false, Bm, (short)0, Cm, false, false);
    // lane n<16 : sum_{m=0..7} rowsum(m); lane n>=16 : sum_{m=8..15} rowsum(m)
    float s = Dm[0] + Dm[1] + Dm[2] + Dm[3] + Dm[4] + Dm[5] + Dm[6] + Dm[7];
    s += __shfl_xor(s, 16, 32);         // full wave sum in every lane

    __shared__ float wsum[8];
    int wid = threadIdx.x >> 5;
    if ((threadIdx.x & 31) == 0) wsum[wid] = s;
    __syncthreads();
    if (threadIdx.x == 0) {
        float t = 0.0f;
        int nw = blockDim.x >> 5;
        for (int i = 0; i < nw; i++) t += wsum[i];
        atomicAdd(&ws[slot], t);
    }
}

// One block per (b, scale): dedupe GT cells, build class bitmasks, evaluate
// the positive-cell obj terms and the masked cls BCE.
__global__ void k_poscells(const float* __restrict__ p0,
                           const float* __restrict__ p1,
                           const float* __restrict__ p2,
                           const float* __restrict__ boxes,
                           const int*   __restrict__ labels,
                           float* ws) {
    int blk = blockIdx.x;        // 0..47
    int b = blk & 15;
    int s = blk >> 4;
    const float* pred = (s == 0) ? p0 : (s == 1) ? p1 : p2;
    int w  = (s == 0) ? 80 : (s == 1) ? 40 : 20;
    int h  = w;
    int hw = w * h;

    __shared__ int      s_cid[NUM_M];
    __shared__ int      s_lab[NUM_M];
    __shared__ int      s_valid[NUM_M];
    __shared__ int      s_nl;
    __shared__ int      s_list[NUM_M];
    __shared__ unsigned s_cm[NUM_M][3];
    __shared__ float    s_acc[3];

    int tid = threadIdx.x;
    if (tid == 0) { s_nl = 0; s_acc[0] = s_acc[1] = s_acc[2] = 0.0f; }
    __syncthreads();

    if (tid < NUM_M) {
        int m = tid;
        const float* bx = boxes + (size_t)(b * NUM_M + m) * 4;
        float x0 = bx[0], y0 = bx[1], x1 = bx[2], y1 = bx[3];
        int lab = labels[b * NUM_M + m];
        int valid = (lab >= 0 && lab < NUM_C) ? 1 : 0;
        float cx = fminf(fmaxf((x0 + x1) * 0.5f / IMG, 0.0f), 1.0f - 1e-6f);
        float cy = fminf(fmaxf((y0 + y1) * 0.5f / IMG, 0.0f), 1.0f - 1e-6f);
        float gx = cx * (float)w, gy = cy * (float)h;
        int gi = min(max((int)floorf(gx), 0), w - 1);
        int gj = min(max((int)floorf(gy), 0), h - 1);
        s_cid[m]   = gj * w + gi;
        s_lab[m]   = min(max(lab, 0), NUM_C - 1);
        s_valid[m] = valid;
    }
    __syncthreads();

    if (tid < NUM_M) {
        int m = tid;
        if (s_valid[m]) {
            bool leader = true;
            for (int mm = 0; mm < m; mm++)
                if (s_valid[mm] && s_cid[mm] == s_cid[m]) { leader = false; break; }
            if (leader) {
                unsigned c0 = 0, c1 = 0, c2 = 0;
                for (int mm = 0; mm < NUM_M; mm++) {
                    if (s_valid[mm] && s_cid[mm] == s_cid[m]) {
                        int c = s_lab[mm];
                        if (c < 32)      c0 |= 1u << c;
                        else if (c < 64) c1 |= 1u << (c - 32);
                        else             c2 |= 1u << (c - 64);
                    }
                }
                int pos = atomicAdd(&s_nl, 1);
                s_list[pos]  = s_cid[m];
                s_cm[pos][0] = c0; s_cm[pos][1] = c1; s_cm[pos][2] = c2;
            }
        }
    }
    __syncthreads();

    int nl = s_nl;
    if (tid == 0) {
        atomicAdd(&ws[12 + s], (float)nl);
        if (s == 0) {
            int cnt = 0;
            for (int m = 0; m < NUM_M; m++) cnt += s_valid[m];
            ws[16 + b] = (float)cnt;   // unique writer per b
        }
    }

    // Tasks over (leader cell, anchor, [obj | class 0..79]) = nl * 3 * 81
    float a_pos1 = 0.0f, a_psub = 0.0f, a_cls = 0.0f;
    int tasks = nl * NUM_A * (NUM_C + 1);
    for (int t = tid; t < tasks; t += blockDim.x) {
        int li = t / (NUM_A * (NUM_C + 1));
        int r  = t - li * (NUM_A * (NUM_C + 1));
        int a  = r / (NUM_C + 1);
        int q  = r - a * (NUM_C + 1);
        int cell = s_list[li];
        const float* base = pred + (size_t)(b * 255 + a * 85) * hw;
        if (q == 0) {
            float x  = base[(size_t)4 * hw + cell];
            float sp = softplusf(x);
            a_pos1 += sp - x;          // BCE(x, 1)
            a_psub += sp;              // BCE(x, 0) to subtract from S0
        } else {
            int c = q - 1;
            float x  = base[(size_t)(5 + c) * hw + cell];
            unsigned bit = (s_cm[li][c >> 5] >> (c & 31)) & 1u;
            a_cls += softplusf(x) - (bit ? x : 0.0f);
        }
    }
    for (int off = 16; off > 0; off >>= 1) {
        a_pos1 += __shfl_xor(a_pos1, off, 32);
        a_psub += __shfl_xor(a_psub, off, 32);
        a_cls  += __shfl_xor(a_cls,  off, 32);
    }
    if ((tid & 31) == 0) {
        atomicAdd(&s_acc[0], a_pos1);
        atomicAdd(&s_acc[1], a_psub);
        atomicAdd(&s_acc[2], a_cls);
    }
    __syncthreads();
    if (tid == 0) {
        atomicAdd(&ws[3 + s], s_acc[0]);
        atomicAdd(&ws[6 + s], s_acc[1]);
        atomicAdd(&ws[9 + s], s_acc[2]);
    }
}

// Box regression: gather 4 coords per (b,m,s,a), sigmoid + smooth-L1.
__global__ void k_box(const float* __restrict__ p0,
                      const float* __restrict__ p1,
                      const float* __restrict__ p2,
                      const float* __restrict__ boxes,
                      const int*   __restrict__ labels,
                      float* ws) {
    const int total = NUM_B * NUM_M * 3 * NUM_A * 4;   // 18432
    float acc = 0.0f;
    for (int t = blockIdx.x * blockDim.x + threadIdx.x; t < total;
         t += gridDim.x * blockDim.x) {
        int k = t & 3;
        int r = t >> 2;
        int a = r % NUM_A; r /= NUM_A;
        int s = r % 3;     r /= 3;
        int m = r % NUM_M;
        int b = r / NUM_M;

        int lab = labels[b * NUM_M + m];
        if (lab < 0 || lab >= NUM_C) continue;

        const float* pred = (s == 0) ? p0 : (s == 1) ? p1 : p2;
        int w  = (s == 0) ? 80 : (s == 1) ? 40 : 20;
        int h  = w;
        int hw = w * h;

        const float* bx = boxes + (size_t)(b * NUM_M + m) * 4;
        float x0 = bx[0], y0 = bx[1], x1 = bx[2], y1 = bx[3];
        float cx = fminf(fmaxf((x0 + x1) * 0.5f / IMG, 0.0f), 1.0f - 1e-6f);
        float cy = fminf(fmaxf((y0 + y1) * 0.5f / IMG, 0.0f), 1.0f - 1e-6f);
        float bw = fminf(fmaxf((x1 - x0) / IMG, 1e-6f), 1.0f);
        float bh = fminf(fmaxf((y1 - y0) / IMG, 1e-6f), 1.0f);
        float gx = cx * (float)w, gy = cy * (float)h;
        int gi = min(max((int)floorf(gx), 0), w - 1);
        int gj = min(max((int)floorf(gy), 0), h - 1);
        float tx = gx - (float)gi, ty = gy - (float)gj;
        float tgt = (k == 0) ? tx : (k == 1) ? ty : (k == 2) ? bw : bh;

        float x = pred[(size_t)(b * 255 + a * 85 + k) * hw + gj * w + gi];
        float p = 1.0f / (1.0f + expf(-x));
        float d = p - tgt;
        float ad = fabsf(d);
        float el = (ad < 1.0f) ? 0.5f * d * d : ad - 0.5f;

        float nv    = ws[16 + b];
        float denom = fmaxf(nv * 4.0f, 1.0f);
        acc += el / denom;
    }
    for (int off = 16; off > 0; off >>= 1) acc += __shfl_xor(acc, off, 32);
    if ((threadIdx.x & 31) == 0) atomicAdd(&ws[15], acc);
}

__global__ void k_final(const float* __restrict__ ws, float* out) {
    if (blockIdx.x == 0 && threadIdx.x == 0) {
        const float hws[3] = {6400.0f, 1600.0f, 400.0f};
        float obj = 0.0f, cls = 0.0f;
        for (int s = 0; s < 3; s++) {
            float pos_cnt = ws[12 + s] * (float)NUM_A;
            float tot     = (float)NUM_B * (float)NUM_A * hws[s];
            float neg_cnt = tot - pos_cnt;
            float obj_pos = (pos_cnt > 0.0f) ? ws[3 + s] / fmaxf(pos_cnt, 1.0f) : 0.0f;
            float obj_neg = (neg_cnt > 0.0f) ? (ws[0 + s] - ws[6 + s]) / fmaxf(neg_cnt, 1.0f)
                                             : 0.0f;
            obj += obj_pos + 0.1f * obj_neg;
            float cls_den = pos_cnt * (float)NUM_C;
            cls += (cls_den > 0.0f) ? ws[9 + s] / fmaxf(cls_den, 1.0f) : 0.0f;
        }
        float nvsum = 0.0f;
        for (int b = 0; b < NUM_B; b++) nvsum += ws[16 + b];
        float total_pos = nvsum * 9.0f;   // 3 scales * A anchors * sum(nv)
        float box_loss  = (total_pos > 0.0f) ? ws[15] / fmaxf(total_pos, 1.0f) : 0.0f;
        out[0] = (obj / 3.0f) + (cls / 3.0f) + 5.0f * box_loss;
    }
}

extern "C" void kernel_launch(void* const* d_in, const int* in_sizes, int n_in,
                              void* d_out, int out_size, void* d_ws, size_t ws_size,
                              hipStream_t stream) {
    const float* p0     = (const float*)d_in[0];
    const float* p1     = (const float*)d_in[1];
    const float* p2     = (const float*)d_in[2];
    const float* boxes  = (const float*)d_in[3];
    const int*   labels = (const int*)d_in[4];
    float* out = (float*)d_out;
    float* ws  = (float*)d_ws;

    k_zero<<<1, 32, 0, stream>>>(ws);
    k_obj<<<300, 256, 0, stream>>>(p0, 6400, ws, 0);
    k_obj<<<75,  256, 0, stream>>>(p1, 1600, ws, 1);
    k_obj<<<19,  256, 0, stream>>>(p2, 400,  ws, 2);
    k_poscells<<<48, 128, 0, stream>>>(p0, p1, p2, boxes, labels, ws);
    k_box<<<72, 256, 0, stream>>>(p0, p1, p2, boxes, labels, ws);
    k_final<<<1, 32, 0, stream>>>(ws, out);
}